// ReconStep_58025008169121
// MI455X (gfx1250) — compile-verified
//
#include <hip/hip_runtime.h>
#include <math.h>

#define NSAMP 96
#define NVOX  (128*128*128)
#define BLK   256

// ---- gfx1250 async global->LDS path (probe-verified) ----------------------
#if defined(__AMDGCN__) && __has_builtin(__builtin_amdgcn_global_load_async_to_lds_b64)
#define USE_ASYNC_LDS 1
typedef int v2i __attribute__((vector_size(8)));
typedef __attribute__((address_space(1))) v2i* gv2i_p;   // global
typedef __attribute__((address_space(3))) v2i* lv2i_p;   // LDS
#else
#define USE_ASYNC_LDS 0
#endif

__device__ __forceinline__ void atomic_add_f32(float* p, float v) {
#if defined(__AMDGCN__)
    unsafeAtomicAdd(p, v);   // global_atomic_add_f32 (no-return form, STOREcnt)
#else
    atomicAdd(p, v);
#endif
}

__global__ __launch_bounds__(BLK) void zero_acc(float* __restrict__ acc) {
    int i = blockIdx.x * BLK + threadIdx.x;
    acc[i] = 0.0f;
}

// One thread = one LOR. Slab-clip the ray to the volume, forward-project the
// live samples (trilinear gathers, image L2-resident), then backproject
// proj*step*w into the 8 corner voxels with hw fp32 atomics.
__global__ __launch_bounds__(BLK) void fused_fwd_bwd(
    const float* __restrict__ img,
    const float* __restrict__ xl,
    const float* __restrict__ yl,
    const float* __restrict__ zl,
    float* __restrict__ acc)
{
    __shared__ float slor[BLK * 6];
    const float* lors = (blockIdx.y == 0) ? xl : (blockIdx.y == 1 ? yl : zl);
    const int t = threadIdx.x;
    const float* lorBase = lors + (size_t)blockIdx.x * (BLK * 6);

    // ---- stage this block's 6 KB of LOR data into LDS via async DMA ------
#if USE_ASYNC_LDS
    {
        const v2i* g = (const v2i*)lorBase;   // 768 qwords per block
        v2i*       l = (v2i*)slor;
        #pragma unroll
        for (int k = 0; k < 3; ++k) {
            int idx = t + k * BLK;            // coalesced per wave
            __builtin_amdgcn_global_load_async_to_lds_b64(
                (gv2i_p)(g + idx), (lv2i_p)(l + idx),
                /*offset=*/0, /*cpol=*/0);
        }
#if __has_builtin(__builtin_amdgcn_s_wait_asynccnt)
        __builtin_amdgcn_s_wait_asynccnt(0);
#else
        asm volatile("s_wait_asynccnt 0" ::: "memory");
#endif
    }
#else
    {
        const unsigned long long* g = (const unsigned long long*)lorBase;
        unsigned long long*       l = (unsigned long long*)slor;
        #pragma unroll
        for (int k = 0; k < 3; ++k) { int idx = t + k * BLK; l[idx] = g[idx]; }
    }
#endif
    __syncthreads();

    const float p1x = slor[t*6+0], p1y = slor[t*6+1], p1z = slor[t*6+2];
    const float p2x = slor[t*6+3], p2y = slor[t*6+4], p2z = slor[t*6+5];
    const float dx = p2x - p1x, dy = p2y - p1y, dz = p2z - p1z;
    const float inv = 1.0f / (float)NSAMP;
    const float step = sqrtf(dx*dx + dy*dy + dz*dz) * inv;

    // vox(s) = o + (s+0.5)*dd   (voxel space; VOXEL=2, CENTER=0, SIZE=256)
    const float ox = (p1x + 128.0f) * 0.5f - 0.5f;
    const float oy = (p1y + 128.0f) * 0.5f - 0.5f;
    const float oz = (p1z + 128.0f) * 0.5f - 0.5f;
    const float ddx = dx * 0.5f * inv;
    const float ddy = dy * 0.5f * inv;
    const float ddz = dz * 0.5f * inv;

    // ---- conservative slab clip: live sample interval [s0, s1) -----------
    // sf = s + 0.5 must satisfy  0 <= o + sf*dd <= 127  on all three axes.
    float lo = 0.0f, hi = (float)NSAMP;
    bool empty = false;
    {
        const float o[3]  = {ox, oy, oz};
        const float dd[3] = {ddx, ddy, ddz};
        #pragma unroll
        for (int a = 0; a < 3; ++a) {
            if (fabsf(dd[a]) > 1e-6f) {
                const float r  = 1.0f / dd[a];
                const float t0 = (0.0f   - o[a]) * r;
                const float t1 = (127.0f - o[a]) * r;
                lo = fmaxf(lo, fminf(t0, t1));
                hi = fminf(hi, fmaxf(t0, t1));
            } else if (o[a] < -1e-3f || o[a] > 127.0f + 1e-3f) {
                empty = true;   // axis ~constant and clearly outside
            }
        }
    }
    int s0 = 0, s1 = 0;
    if (!empty && hi > lo) {
        s0 = (int)fmaxf(0.0f, floorf(lo - 0.5f)) - 1;      // widen 1 sample
        s1 = (int)fminf((float)NSAMP, ceilf(hi - 0.5f) + 1.0f) + 1;
        s0 = s0 < 0 ? 0 : s0;
        s1 = s1 > NSAMP ? NSAMP : s1;
    }

    // ---- forward: line integral over live samples ------------------------
    float proj = 0.0f;
    #pragma unroll 2
    for (int s = s0; s < s1; ++s) {
        const float ts = (float)s + 0.5f;
        const float fx = ox + ts * ddx;
        const float fy = oy + ts * ddy;
        const float fz = oz + ts * ddz;
        const bool inb = (fx >= 0.0f) & (fx <= 127.0f) &
                         (fy >= 0.0f) & (fy <= 127.0f) &
                         (fz >= 0.0f) & (fz <= 127.0f);
        if (inb) {
            const int ix = (int)fx, iy = (int)fy, iz = (int)fz;
            const float ax = fx - (float)ix, ay = fy - (float)iy, az = fz - (float)iz;
            const float bx = 1.0f - ax,      by = 1.0f - ay,      bz = 1.0f - az;
            const int base = (ix << 14) | (iy << 7) | iz;
            const int sx = (ix < 127) ? (1 << 14) : 0;
            const int sy = (iy < 127) ? (1 << 7)  : 0;
            const int sz = (iz < 127) ? 1         : 0;
            const float v000 = img[base],           v001 = img[base + sz];
            const float v010 = img[base + sy],      v011 = img[base + sy + sz];
            const float v100 = img[base + sx],      v101 = img[base + sx + sz];
            const float v110 = img[base + sx + sy], v111 = img[base + sx + sy + sz];
            proj += bx * (by * (bz * v000 + az * v001) + ay * (bz * v010 + az * v011))
                  + ax * (by * (bz * v100 + az * v101) + ay * (bz * v110 + az * v111));
        }
    }
    proj *= step;
    const float c = proj * step;   // VJP cotangent (= proj) times step chain

    // ---- backward: scatter c * w into the 8 corners ----------------------
    if (c != 0.0f) {
        for (int s = s0; s < s1; ++s) {
            const float ts = (float)s + 0.5f;
            const float fx = ox + ts * ddx;
            const float fy = oy + ts * ddy;
            const float fz = oz + ts * ddz;
            const bool inb = (fx >= 0.0f) & (fx <= 127.0f) &
                             (fy >= 0.0f) & (fy <= 127.0f) &
                             (fz >= 0.0f) & (fz <= 127.0f);
            if (inb) {
                const int ix = (int)fx, iy = (int)fy, iz = (int)fz;
                const float ax = fx - (float)ix, ay = fy - (float)iy, az = fz - (float)iz;
                const float bx = 1.0f - ax,      by = 1.0f - ay,      bz = 1.0f - az;
                const int base = (ix << 14) | (iy << 7) | iz;
                const int sx = (ix < 127) ? (1 << 14) : 0;
                const int sy = (iy < 127) ? (1 << 7)  : 0;
                const int sz = (iz < 127) ? 1         : 0;
                const float cbx = c * bx, cax = c * ax;
                atomic_add_f32(&acc[base],                cbx * by * bz);
                atomic_add_f32(&acc[base + sz],           cbx * by * az);
                atomic_add_f32(&acc[base + sy],           cbx * ay * bz);
                atomic_add_f32(&acc[base + sy + sz],      cbx * ay * az);
                atomic_add_f32(&acc[base + sx],           cax * by * bz);
                atomic_add_f32(&acc[base + sx + sz],      cax * by * az);
                atomic_add_f32(&acc[base + sx + sy],      cax * ay * bz);
                atomic_add_f32(&acc[base + sx + sy + sz], cax * ay * az);
            }
        }
    }
}

__global__ __launch_bounds__(BLK) void finalize_mlem(
    const float* __restrict__ img,
    const float* __restrict__ eff,
    float* __restrict__ out)   // out currently holds the backprojection
{
    int i = blockIdx.x * BLK + threadIdx.x;
    out[i] = img[i] / (eff[i] + 1e-8f) * out[i];
}

extern "C" void kernel_launch(void* const* d_in, const int* in_sizes, int n_in,
                              void* d_out, int out_size, void* d_ws, size_t ws_size,
                              hipStream_t stream) {
    const float* img = (const float*)d_in[0];
    const float* eff = (const float*)d_in[1];
    const float* xl  = (const float*)d_in[2];
    const float* yl  = (const float*)d_in[3];
    const float* zl  = (const float*)d_in[4];
    float* out = (float*)d_out;

    const int nlors   = in_sizes[2] / 6;      // 65536 per set
    const int lblocks = nlors / BLK;          // 256
    const int vblocks = NVOX / BLK;           // 8192

    // d_out doubles as the backprojection accumulator: zero -> scatter -> scale.
    hipLaunchKernelGGL(zero_acc,      dim3(vblocks),    dim3(BLK), 0, stream, out);
    hipLaunchKernelGGL(fused_fwd_bwd, dim3(lblocks, 3), dim3(BLK), 0, stream,
                       img, xl, yl, zl, out);
    hipLaunchKernelGGL(finalize_mlem, dim3(vblocks),    dim3(BLK), 0, stream,
                       img, eff, out);
}